// VectorQuantization_45621142618435
// MI455X (gfx1250) — compile-verified
//
#include <hip/hip_runtime.h>
#include <stdint.h>

#define DIM        64
#define K_EMBED    1024
#define N_ROWS     (2048 * 128)
#define DECAYF     0.99f
#define OMDECAYF   0.01f
#define EPSF       1e-5f

typedef __attribute__((ext_vector_type(2))) float v2f;
typedef __attribute__((ext_vector_type(8))) float v8f;

// ---- workspace layout (in floats) ----
#define WS_SQ      0          // 1: sum of squared errors over valid rows
#define WS_NV      1          // 1: number of valid rows
#define WS_TOTAL   2          // 1: sum of new_cluster_size
#define WS_COUNTS  64         // 1024: per-code counts
#define WS_ESUM    1088       // 64*1024: embed_sum[d][k]
#define WS_HNORM   66624      // 1024: 0.5*||e_k||^2
#define WS_IND     67648      // 262144 int32: argmin indices

// ---- gfx1250 async global->LDS (guarded; falls back to sync copy) ----
#if defined(__has_builtin)
#if __has_builtin(__builtin_amdgcn_global_load_async_to_lds_b32)
#define VQ_HAVE_ASYNC 1
#endif
#endif

__device__ __forceinline__ void vq_stage_b32(const float* g, float* l) {
#if defined(VQ_HAVE_ASYNC)
    typedef __attribute__((address_space(1))) int gi_t;
    typedef __attribute__((address_space(3))) int li_t;
    // generic LDS pointer low 32 bits == LDS offset (flat aperture rule)
    __builtin_amdgcn_global_load_async_to_lds_b32(
        (gi_t*)(uintptr_t)g, (li_t*)(unsigned)(uintptr_t)l, 0, 0);
#else
    *l = *g;
#endif
}

__device__ __forceinline__ void vq_wait_async() {
#if defined(VQ_HAVE_ASYNC)
#if __has_builtin(__builtin_amdgcn_s_wait_asynccnt)
    __builtin_amdgcn_s_wait_asynccnt(0);
#else
    asm volatile("s_wait_asynccnt 0" ::: "memory");
#endif
#endif
}

// ---------------------------------------------------------------------------
// Kernel 1: half squared norms of codebook columns
// ---------------------------------------------------------------------------
__global__ void vq_halfnorm(const float* __restrict__ embed,
                            float* __restrict__ hnorm) {
    int k = blockIdx.x * blockDim.x + threadIdx.x;
    if (k < K_EMBED) {
        float s = 0.f;
#pragma unroll
        for (int d = 0; d < DIM; ++d) {
            float e = embed[d * K_EMBED + k];
            s += e * e;
        }
        hnorm[k] = 0.5f * s;
    }
}

// ---------------------------------------------------------------------------
// Kernel 2: WMMA argmin. Block = 256 threads = 8 waves, each wave owns a
// 16-row tile. Codebook is staged transposed into LDS in 128-column chunks
// (row stride 66 floats: conflict-free banks + 8B alignment for b64 loads),
// via async global->LDS loads when available.
// score[n][k] = x_n . e_k - 0.5*||e_k||^2 ; argmax == reference argmin.
// ---------------------------------------------------------------------------
__global__ __launch_bounds__(256) void vq_argmin(const float* __restrict__ x,
                                                 const float* __restrict__ embed,
                                                 const float* __restrict__ hnorm,
                                                 int* __restrict__ ind) {
    __shared__ float lds[128 * 66];   // 33 KB: 128 codes x 64 dims (stride 66)

    const int tid     = threadIdx.x;
    const int lane    = tid & 31;
    const int wave    = tid >> 5;
    const int halfsel = lane >> 4;    // 0: lanes 0-15, 1: lanes 16-31
    const int lrow    = lane & 15;
    const int rowbase = blockIdx.x * 128 + wave * 16;

    // A operand in registers, V_WMMA_F32_16X16X4_F32 layout:
    // a[c] covers depth chunk K = 4c .. 4c+3 ; lane half selects K pair.
    v2f a[16];
#pragma unroll
    for (int c = 0; c < 16; ++c)
        a[c] = *(const v2f*)&x[(rowbase + lrow) * DIM + 4 * c + 2 * halfsel];

    float best[8];
    int   bidx[8];
#pragma unroll
    for (int s = 0; s < 8; ++s) { best[s] = -3.0e38f; bidx[s] = 0; }

    for (int cg = 0; cg < 8; ++cg) {           // 8 column groups of 128 codes
        __syncthreads();
        // transpose-stage: embed[d][k] -> lds[k_local*66 + d]
#pragma unroll
        for (int i = 0; i < 32; ++i) {
            int idx = tid + 256 * i;           // 8192 = 128 cols * 64 dims
            int c = idx & 127;
            int d = idx >> 7;
            vq_stage_b32(&embed[d * K_EMBED + cg * 128 + c], &lds[c * 66 + d]);
        }
        vq_wait_async();
        __syncthreads();

        for (int t = 0; t < 8; ++t) {          // 8 tiles of 16 columns
            // Preload the full B tile first: lets all DS loads issue together.
            const float* bp = &lds[(t * 16 + lrow) * 66 + 2 * halfsel];
            v2f b[16];
#pragma unroll
            for (int c = 0; c < 16; ++c)
                b[c] = *(const v2f*)(bp + 4 * c);

            // Two independent accumulator chains for WMMA ILP.
            v8f acc0 = {}, acc1 = {};
#pragma unroll
            for (int c = 0; c < 8; ++c) {
                acc0 = __builtin_amdgcn_wmma_f32_16x16x4_f32(
                        false, a[2 * c], false, b[2 * c],
                        (short)0, acc0, false, false);
                acc1 = __builtin_amdgcn_wmma_f32_16x16x4_f32(
                        false, a[2 * c + 1], false, b[2 * c + 1],
                        (short)0, acc1, false, false);
            }

            const int   mycol = cg * 128 + t * 16 + lrow;
            const float hn    = hnorm[mycol];
#pragma unroll
            for (int s = 0; s < 8; ++s) {
                float v = (acc0[s] + acc1[s]) - hn;
                if (v > best[s]) { best[s] = v; bidx[s] = mycol; }   // strict >
            }                                   // keeps lowest-index ties
        }
    }

    // Cross-lane argmax within each 16-lane group (same row per C slot).
#pragma unroll
    for (int s = 0; s < 8; ++s) {
        float v  = best[s];
        int   bi = bidx[s];
#pragma unroll
        for (int off = 8; off >= 1; off >>= 1) {
            float ov = __shfl_xor(v, off, 16);
            int   oi = __shfl_xor(bi, off, 16);
            if (ov > v || (ov == v && oi < bi)) { v = ov; bi = oi; }
        }
        best[s] = v; bidx[s] = bi;
    }
    if (lrow == 0) {
#pragma unroll
        for (int s = 0; s < 8; ++s)
            ind[rowbase + halfsel * 8 + s] = bidx[s];   // rows 0-7 / 8-15
    }
}

// ---------------------------------------------------------------------------
// Kernel 3: gather quantize output, masked MSE partials, scatter counts and
// embed_sum. One wave per row (2 dims per lane).
// ---------------------------------------------------------------------------
__global__ __launch_bounds__(256) void vq_gather(const float* __restrict__ x,
                                                 const unsigned char* __restrict__ mask,
                                                 const float* __restrict__ embed,
                                                 const int* __restrict__ ind,
                                                 float* __restrict__ quant,
                                                 float* __restrict__ counts,
                                                 float* __restrict__ esum,
                                                 float* __restrict__ sq_acc,
                                                 float* __restrict__ nv_acc) {
    const int lane = threadIdx.x & 31;
    const int wave = threadIdx.x >> 5;
    const int n    = blockIdx.x * 8 + wave;
    const int k    = ind[n];
    const int d0   = lane * 2;

    v2f xv = *(const v2f*)&x[n * DIM + d0];
    float e0 = embed[d0 * K_EMBED + k];
    float e1 = embed[(d0 + 1) * K_EMBED + k];

    v2f q; q.x = e0; q.y = e1;
    *(v2f*)&quant[n * DIM + d0] = q;           // straight-through value == code

    const bool valid = (mask[n] == 0);
    if (valid) {
        atomicAdd(&esum[d0 * K_EMBED + k],       xv.x);
        atomicAdd(&esum[(d0 + 1) * K_EMBED + k], xv.y);
    }

    float dx = e0 - xv.x, dy = e1 - xv.y;
    float s = dx * dx + dy * dy;
#pragma unroll
    for (int off = 16; off >= 1; off >>= 1) s += __shfl_xor(s, off, 32);

    if (lane == 0 && valid) {
        atomicAdd(sq_acc, s);
        atomicAdd(nv_acc, 1.0f);
        atomicAdd(&counts[k], 1.0f);
    }
}

// ---------------------------------------------------------------------------
// Kernel 4: new_cluster_size + total
// ---------------------------------------------------------------------------
__global__ void vq_cluster(const float* __restrict__ cs,
                           const float* __restrict__ counts,
                           float* __restrict__ ncs_out,
                           float* __restrict__ total) {
    int k = blockIdx.x * blockDim.x + threadIdx.x;
    if (k < K_EMBED) {
        float v = cs[k] * DECAYF + OMDECAYF * counts[k];
        ncs_out[k] = v;
        atomicAdd(total, v);
    }
}

// ---------------------------------------------------------------------------
// Kernel 5: new_embed_avg, smoothed new_embed, loss scalar
// ---------------------------------------------------------------------------
__global__ void vq_update(const float* __restrict__ eavg,
                          const float* __restrict__ esum,
                          const float* __restrict__ ncs,
                          const float* __restrict__ total_p,
                          const float* __restrict__ sq,
                          const float* __restrict__ nv,
                          float* __restrict__ new_embed,
                          float* __restrict__ new_eavg,
                          float* __restrict__ loss_out) {
    int i = blockIdx.x * blockDim.x + threadIdx.x;
    if (i == 0)
        loss_out[0] = sq[0] / (nv[0] * (float)DIM);
    if (i < DIM * K_EMBED) {
        int   k     = i & (K_EMBED - 1);
        float nea   = eavg[i] * DECAYF + OMDECAYF * esum[i];
        new_eavg[i] = nea;
        float total = total_p[0];
        float sm    = (ncs[k] + EPSF) / (total + (float)K_EMBED * EPSF) * total;
        new_embed[i] = nea / sm;
    }
}

// ---------------------------------------------------------------------------
extern "C" void kernel_launch(void* const* d_in, const int* in_sizes, int n_in,
                              void* d_out, int out_size, void* d_ws, size_t ws_size,
                              hipStream_t stream) {
    const float*         x     = (const float*)d_in[0];
    const unsigned char* mask  = (const unsigned char*)d_in[1]; // bool array
    const float*         embed = (const float*)d_in[2];
    const float*         cs    = (const float*)d_in[3];
    const float*         eavg  = (const float*)d_in[4];

    float* out    = (float*)d_out;
    float* quant  = out;                         // 2048*128*64
    float* loss   = out + (size_t)N_ROWS * DIM;  // 1
    float* nembed = loss + 1;                    // 64*1024
    float* ncs    = nembed + DIM * K_EMBED;      // 1024
    float* neavg  = ncs + K_EMBED;               // 64*1024

    float* ws   = (float*)d_ws;
    int*   indp = (int*)(ws + WS_IND);

    // zero accumulators / counts / embed_sum (everything before WS_HNORM)
    (void)hipMemsetAsync(d_ws, 0, (size_t)WS_HNORM * sizeof(float), stream);

    vq_halfnorm<<<4, 256, 0, stream>>>(embed, ws + WS_HNORM);

    vq_argmin<<<N_ROWS / 128, 256, 0, stream>>>(x, embed, ws + WS_HNORM, indp);

    vq_gather<<<N_ROWS / 8, 256, 0, stream>>>(x, mask, embed, indp, quant,
                                              ws + WS_COUNTS, ws + WS_ESUM,
                                              ws + WS_SQ, ws + WS_NV);

    vq_cluster<<<4, 256, 0, stream>>>(cs, ws + WS_COUNTS, ncs, ws + WS_TOTAL);

    vq_update<<<(DIM * K_EMBED + 255) / 256, 256, 0, stream>>>(
        eavg, ws + WS_ESUM, ncs, ws + WS_TOTAL, ws + WS_SQ, ws + WS_NV,
        nembed, neavg, loss);
}